// LocalViKeyPointBottleneck_24670292148447
// MI455X (gfx1250) — compile-verified
//
#include <hip/hip_runtime.h>
#include <hip/hip_bf16.h>
#include <math.h>

typedef __attribute__((ext_vector_type(2))) float v2f;
typedef __attribute__((ext_vector_type(8))) float v8f;

#define EPS_BN 1e-5f

// ---------------------------------------------------------------------------
// Repack conv weights (3,3,Cin=64,Cout) HWIO -> WMMA A-fragment order:
//   pair index = ((tap*OCT + ot)*32 + icp)*16 + col   (float2 = ic even/odd)
// so a wave's K-step A fetch is 16 contiguous float2 per half-wave (b64 loads).
// ---------------------------------------------------------------------------
__global__ void repack_convw_kernel(const float* __restrict__ w,
                                    float* __restrict__ wp, int Cout)
{
    int OCT = Cout >> 4;
    int total = 9 * OCT * 32 * 16;
    int idx = blockIdx.x * blockDim.x + threadIdx.x;
    if (idx >= total) return;
    int col = idx & 15;
    int r   = idx >> 4;
    int icp = r % 32;  r /= 32;
    int ot  = r % OCT; int tap = r / OCT;
    int oc  = ot * 16 + col;
    wp[2 * idx + 0] = w[((size_t)tap * 64 + icp * 2 + 0) * Cout + oc];
    wp[2 * idx + 1] = w[((size_t)tap * 64 + icp * 2 + 1) * Cout + oc];
}

// wl (400,128) -> pair index = kp*128 + d  (float2 = k even/odd)
__global__ void repack_wl_kernel(const float* __restrict__ wl,
                                 float* __restrict__ wlp)
{
    int idx = blockIdx.x * blockDim.x + threadIdx.x;
    if (idx >= 200 * 128) return;
    int d = idx & 127, kp = idx >> 7;
    wlp[2 * idx + 0] = wl[(size_t)(2 * kp + 0) * 128 + d];
    wlp[2 * idx + 1] = wl[(size_t)(2 * kp + 1) * 128 + d];
}

// ---------------------------------------------------------------------------
// Kernel 1: transposed conv (3x3, stride 2, VALID) + bias + BN + ReLU + scale
// Implicit GEMM, V_WMMA_F32_16X16X4_F32.
//   block: (32, Cout/16) -> one wave per 16-oc tile, 16 output columns/tile
// Input tile (<=2 rows x 64ch x 10 cols) staged to LDS once per block via
// async global->LDS (CDNA5), shared by all waves; B read as ds_load_b64.
// ---------------------------------------------------------------------------
__global__ void convt_bn_relu_kernel(
    const float* __restrict__ in,    // (N, 64, Hin, Win)
    const float* __restrict__ wpack, // repacked weights
    const float* __restrict__ bias,
    const float* __restrict__ bng,
    const float* __restrict__ bnb,
    const float* __restrict__ bnm,
    const float* __restrict__ bnv,
    float* __restrict__ out,         // (N, Cout, Hout, Wout)
    int N, int Hin, int Win,
    int Cout, int Hout, int Wout,
    float outscale)
{
    const int Cin = 64;
    const int OCT = Cout >> 4;
    const int wtiles = (Wout + 15) >> 4;

    int bid = blockIdx.x;
    int wt  = bid % wtiles;
    int t2  = bid / wtiles;
    int oh  = t2 % Hout;
    int n   = t2 / Hout;

    int ow0 = wt * 16;
    int ot  = threadIdx.y;           // oc tile for this wave
    int oc0 = ot * 16;

    int lane = threadIdx.x;          // 0..31
    int col  = lane & 15;
    int hi   = lane >> 4;
    int ow   = ow0 + col;

    // valid (kh, ih) rows for this output row (uniform over block): <= 2
    int khs[2], ihs[2], nrows = 0;
    for (int kh = 0; kh < 3; ++kh) {
        int th = oh - kh;
        if (th >= 0 && !(th & 1) && (th >> 1) < Hin) {
            khs[nrows] = kh; ihs[nrows] = th >> 1; ++nrows;
        }
    }

    // Input columns needed: iw in [ow0/2 - 1, ow0/2 + 8]  (10 columns)
    int iw0 = (ow0 >> 1) - 1;

    // LDS: pairs (ic even, ic odd): [row(2)][icp(32)][iw(10)] x float2
    __shared__ float sIn[2 * 32 * 10 * 2];   // 5120 B

    // ---- async stage global -> LDS ----
    {
        int tid  = threadIdx.y * 32 + threadIdx.x;
        int nthr = blockDim.y * 32;
        int tot  = nrows * 1280;             // 1280 floats per row-group
        for (int e = tid; e < tot; e += nthr) {
            int half = e & 1;
            int rest = e >> 1;
            int iw   = rest % 10; rest /= 10;
            int icp  = rest % 32;
            int r    = rest / 32;
            int giw  = iw0 + iw;
            if (giw >= 0 && giw < Win) {
                const float* gsrc = in +
                    ((size_t)(n * Cin + icp * 2 + half) * Hin + ihs[r]) * Win + giw;
                unsigned lds = (unsigned)(uintptr_t)(&sIn[e]);     // low 32b = LDS addr
                unsigned long long ga = (unsigned long long)(uintptr_t)gsrc;
                asm volatile("global_load_async_to_lds_b32 %0, %1, off"
                             :: "v"(lds), "v"(ga) : "memory");
            }
        }
        asm volatile("s_wait_asynccnt 0x0" ::: "memory");
    }
    __syncthreads();

    const v2f  zero2 = {0.0f, 0.0f};
    const v2f* sB    = (const v2f*)sIn;

    v8f c = {};
    for (int r = 0; r < nrows; ++r) {
        int kh = khs[r];
        for (int kw = 0; kw < 3; ++kw) {
            int  tw    = ow - kw;
            bool bval  = (tw >= 0) && !(tw & 1) && ((tw >> 1) < Win);
            int  iwl   = bval ? ((tw >> 1) - iw0) : 0;       // 0..9
            int  tap   = kh * 3 + kw;
            const v2f* wrow = (const v2f*)wpack +
                              ((size_t)(tap * OCT + ot) * 32) * 16 + col;
            const v2f* brow = sB + (size_t)(r * 32) * 10 + iwl;
#pragma unroll 4
            for (int icp0 = 0; icp0 < 32; icp0 += 2) {
                int icp = icp0 + hi;
                v2f a = wrow[(size_t)icp * 16];              // global_load_b64
                v2f b = brow[(size_t)icp * 10];              // ds_load_b64
                b = bval ? b : zero2;                        // v_cndmask, no branch
                c = __builtin_amdgcn_wmma_f32_16x16x4_f32(
                        false, a, false, b, (short)0, c, false, false);
            }
        }
    }

    // Epilogue: lane holds column 'col', rows m = hi*8 + r
    if (ow < Wout) {
        for (int r = 0; r < 8; ++r) {
            int oc = oc0 + hi * 8 + r;
            float v = c[r] + bias[oc];
            v = bng[oc] * (v - bnm[oc]) * rsqrtf(bnv[oc] + EPS_BN) + bnb[oc];
            v = v > 0.0f ? v : 0.0f;
            v *= outscale;
            out[(((size_t)n * Cout + oc) * Hout + oh) * Wout + ow] = v;
        }
    }
}

// ---------------------------------------------------------------------------
// Kernel 2: soft-argmax coordinates per (n,k) heatmap (259x259 of hm2)
// ---------------------------------------------------------------------------
#define HM 259

__global__ void coord_kernel(const float* __restrict__ hm2,
                             float* __restrict__ mu /* (N*K, 2) */)
{
    int nk = blockIdx.x;
    const float* p = hm2 + (size_t)nk * HM * HM;

    __shared__ float sX[HM];
    __shared__ float sY[HM];
    __shared__ float red[256];

    int t = threadIdx.x;

    for (int w = t; w < HM; w += 256) {
        float s = 0.0f;
        for (int h = 0; h < HM; ++h) s += p[(size_t)h * HM + w];
        sX[w] = s;
    }
    for (int h = t; h < HM; h += 256) {
        float s = 0.0f;
        const float* row = p + (size_t)h * HM;
        for (int w = 0; w < HM; ++w) s += row[w];
        sY[h] = s;
    }
    __syncthreads();

    const float invH = 1.0f / (float)HM;       // softmax of the *mean*
    for (int axis = 0; axis < 2; ++axis) {
        const float* sv = (axis == 0) ? sX : sY;

        float m = -INFINITY;
        for (int i = t; i < HM; i += 256) m = fmaxf(m, sv[i] * invH);
        red[t] = m; __syncthreads();
        for (int s = 128; s > 0; s >>= 1) {
            if (t < s) red[t] = fmaxf(red[t], red[t + s]);
            __syncthreads();
        }
        m = red[0]; __syncthreads();

        float se = 0.0f, sw = 0.0f;
        for (int i = t; i < HM; i += 256) {
            float e = expf(sv[i] * invH - m);
            se += e;
            sw += e * (-1.0f + 2.0f * (float)i / (float)(HM - 1));
        }
        red[t] = se; __syncthreads();
        for (int s = 128; s > 0; s >>= 1) {
            if (t < s) red[t] += red[t + s];
            __syncthreads();
        }
        se = red[0]; __syncthreads();

        red[t] = sw; __syncthreads();
        for (int s = 128; s > 0; s >>= 1) {
            if (t < s) red[t] += red[t + s];
            __syncthreads();
        }
        sw = red[0]; __syncthreads();

        if (t == 0) mu[(size_t)nk * 2 + axis] = sw / se;
    }
}

// ---------------------------------------------------------------------------
// Kernel 3: gaussian maps on the 64x64 grid
// ---------------------------------------------------------------------------
__global__ void gauss_kernel(const float* __restrict__ mu,
                             float* __restrict__ gmaps, int total)
{
    int idx = blockIdx.x * blockDim.x + threadIdx.x;
    if (idx >= total) return;
    int w  = idx & 63;
    int h  = (idx >> 6) & 63;
    int nk = idx >> 12;
    float gx = mu[(size_t)nk * 2 + 0];
    float gy = mu[(size_t)nk * 2 + 1];
    float xs = -1.0f + 2.0f * (float)w / 63.0f;
    float ys = -1.0f + 2.0f * (float)h / 63.0f;
    float dx = xs - gx, dy = ys - gy;
    gmaps[idx] = expf(-(dx * dx + dy * dy) * 100.0f);   // (1/STD)^2
}

// ---------------------------------------------------------------------------
// Kernel 4: 20x20 clamped/masked crop around rounded center (vi_mode_on == 0)
// ---------------------------------------------------------------------------
__global__ void crop_kernel(const float* __restrict__ mu,
                            const float* __restrict__ gmaps,
                            float* __restrict__ cent, int total)
{
    int idx = blockIdx.x * blockDim.x + threadIdx.x;
    if (idx >= total) return;
    int px = idx % 20;
    int py = (idx / 20) % 20;
    int nk = idx / 400;
    float gx = mu[(size_t)nk * 2 + 0];
    float gy = mu[(size_t)nk * 2 + 1];
    int cx = (int)rintf((gx + 1.0f) * 32.0f) - 1;   // round-half-even == jnp.round
    int cy = (int)rintf((gy + 1.0f) * 32.0f) - 1;
    int sx = cx + px - 10;
    int sy = cy + py - 10;
    bool ok = (sx >= 0) && (sx < 64) && (sy >= 0) && (sy < 64);
    int sxc = min(max(sx, 0), 63);
    int syc = min(max(sy, 0), 63);
    cent[idx] = ok ? gmaps[(size_t)nk * 4096 + syc * 64 + sxc] : 0.0f;
}

// ---------------------------------------------------------------------------
// Kernel 5: h = relu(centered(512x400) @ wl(400x128) + bl), WMMA f32,
// packed wl -> contiguous b64 loads for both A and B fragments.
// ---------------------------------------------------------------------------
__global__ void linear_kernel(const float* __restrict__ cent, // 512 x 400
                              const float* __restrict__ wlp,  // packed pairs
                              const float* __restrict__ bl,   // 128
                              float* __restrict__ hout)       // 512 x 128
{
    int lane = threadIdx.x & 31;
    int wave = threadIdx.x >> 5;     // 0..7
    int r0 = blockIdx.x * 16;
    int d0 = wave * 16;
    int col = lane & 15, hi = lane >> 4;
    int d = d0 + col;

    const float* arow = cent + (size_t)(r0 + col) * 400;
    const v2f*   wB   = (const v2f*)wlp;

    v8f c = {};
#pragma unroll 4
    for (int k0 = 0; k0 < 400; k0 += 4) {
        int kp = (k0 >> 1) + hi;                         // k pair index
        v2f a = *(const v2f*)(arow + 2 * kp);            // global_load_b64
        v2f b = wB[(size_t)kp * 128 + d];                // global_load_b64
        c = __builtin_amdgcn_wmma_f32_16x16x4_f32(
                false, a, false, b, (short)0, c, false, false);
    }

    float bv = bl[d];
    for (int r = 0; r < 8; ++r) {
        int row = r0 + hi * 8 + r;
        float v = c[r] + bv;
        hout[(size_t)row * 128 + d] = v > 0.0f ? v : 0.0f;
    }
}

// ---------------------------------------------------------------------------
// Launch
// ---------------------------------------------------------------------------
extern "C" void kernel_launch(void* const* d_in, const int* in_sizes, int n_in,
                              void* d_out, int out_size, void* d_ws, size_t ws_size,
                              hipStream_t stream) {
    const float* x     = (const float*)d_in[0];
    const float* w1    = (const float*)d_in[1];
    const float* b1    = (const float*)d_in[2];
    const float* bn1_g = (const float*)d_in[3];
    const float* bn1_b = (const float*)d_in[4];
    const float* bn1_m = (const float*)d_in[5];
    const float* bn1_v = (const float*)d_in[6];
    const float* w2    = (const float*)d_in[7];
    const float* b2    = (const float*)d_in[8];
    const float* bn2_g = (const float*)d_in[9];
    const float* bn2_b = (const float*)d_in[10];
    const float* bn2_m = (const float*)d_in[11];
    const float* bn2_v = (const float*)d_in[12];
    const float* wl    = (const float*)d_in[13];
    const float* bl    = (const float*)d_in[14];
    // d_in[15] = vi_mode_on (0 in setup_inputs; only that path is shape-consistent)

    float* out = (float*)d_out;
    // Output layout (floats): h | gauss_mu | gauss_maps | centered
    const size_t OFF_MU   = 65536;                 // after h (16*32*128)
    const size_t OFF_GM   = 66560;                 // after gauss_mu (+1024)
    const size_t OFF_CENT = 2163712;               // after gauss_maps (+2097152)

    float* ws  = (float*)d_ws;
    float* h1  = ws;                               // 17,040,384 f
    float* hm2 = ws + 17040384ull;                 // 34,345,472 f
    float* w1p = ws + 51385856ull;                 // 36,864 f
    float* w2p = ws + 51422720ull;                 // 18,432 f
    float* wlp = ws + 51441152ull;                 // 51,200 f

    // weight repacks (WMMA fragment order)
    repack_convw_kernel<<<(18432 + 255) / 256, 256, 0, stream>>>(w1, w1p, 64);
    repack_convw_kernel<<<(9216  + 255) / 256, 256, 0, stream>>>(w2, w2p, 32);
    repack_wl_kernel<<<(25600 + 255) / 256, 256, 0, stream>>>(wl, wlp);

    // convT1 + BN1 + ReLU  -> h1 (16,64,129,129)
    {
        dim3 blk(32, 4);
        int grid = 16 * 129 * ((129 + 15) / 16);
        convt_bn_relu_kernel<<<grid, blk, 0, stream>>>(
            x, w1p, b1, bn1_g, bn1_b, bn1_m, bn1_v, h1,
            16, 64, 64, 64, 129, 129, 1.0f);
    }
    // convT2 + BN2 + ReLU, fused *2  -> hm2 (16,32,259,259)
    {
        dim3 blk(32, 2);
        int grid = 16 * 259 * ((259 + 15) / 16);
        convt_bn_relu_kernel<<<grid, blk, 0, stream>>>(
            h1, w2p, b2, bn2_g, bn2_b, bn2_m, bn2_v, hm2,
            16, 129, 129, 32, 259, 259, 2.0f);
    }
    // soft-argmax -> gauss_mu
    coord_kernel<<<512, 256, 0, stream>>>(hm2, out + OFF_MU);
    // gaussian maps
    gauss_kernel<<<(2097152 + 255) / 256, 256, 0, stream>>>(
        out + OFF_MU, out + OFF_GM, 2097152);
    // crop -> centered
    crop_kernel<<<(204800 + 255) / 256, 256, 0, stream>>>(
        out + OFF_MU, out + OFF_GM, out + OFF_CENT, 204800);
    // final GEMM -> h
    linear_kernel<<<32, 256, 0, stream>>>(out + OFF_CENT, wlp, bl, out);
}